// UpdateOnRightWithSelectiveResetOnLeft_24739011625361
// MI455X (gfx1250) — compile-verified
//
#include <hip/hip_runtime.h>
#include <stdint.h>

// ---------------------------------------------------------------------------
// out[b,n] = [[A1n, 0],[B1n, I]] @ A2_atop_B2[b,n]
//   => out_top = A1n @ X ; out_bot = B1n @ X + Y   (X/Y = top/bottom 32x32)
// Memory-bound (~1.07 GB @ 23.3 TB/s ~= 46us); fp32 WMMA 16x16x4 for compute.
// ---------------------------------------------------------------------------

#define NPB 16          // n-tiles processed per workgroup
#define LDS_STRIDE 44   // dwords per staged row: bank-conflict-free ds reads

typedef __attribute__((ext_vector_type(2))) float v2f;
typedef __attribute__((ext_vector_type(4))) float v4f;
typedef __attribute__((ext_vector_type(8))) float v8f;

// Async global->LDS builtin wants pointers to 16-byte int vectors.
typedef int v4i_vs __attribute__((vector_size(16)));
typedef __attribute__((address_space(1))) v4i_vs* as1_b128;
typedef __attribute__((address_space(3))) v4i_vs* as3_b128;

#if __has_builtin(__builtin_amdgcn_global_load_async_to_lds_b128)
#define HAVE_ASYNC_LDS 1
#else
#define HAVE_ASYNC_LDS 0
#endif

#if HAVE_ASYNC_LDS
#if __has_builtin(__builtin_amdgcn_s_wait_asynccnt)
#define WAIT_ASYNC(n) __builtin_amdgcn_s_wait_asynccnt(n)
#else
#define WAIT_ASYNC(n) asm volatile("s_wait_asynccnt %0" ::"n"(n))
#endif
#else
#define WAIT_ASYNC(n)
#endif

// Stage one 64x32 f32 tile (8KB) global -> LDS, 256 threads, b128 per lane,
// fully coalesced. Async path uses ASYNCcnt-tracked direct-to-LDS loads.
__device__ __forceinline__ void stage_tile(float* lbuf, const float* gsrc, int t) {
#pragma unroll
  for (int it = 0; it < 2; ++it) {
    const int s   = t + 256 * it;       // 512 x 16B segments
    const int row = s >> 3;
    const int seg = s & 7;
    const float* g = gsrc + row * 32 + seg * 4;
    float*       l = lbuf + row * LDS_STRIDE + seg * 4;
#if HAVE_ASYNC_LDS
    __builtin_amdgcn_global_load_async_to_lds_b128(
        (as1_b128)(uintptr_t)g, (as3_b128)(uint32_t)(uintptr_t)l, 0, 0);
#else
    *(v4f*)l = *(const v4f*)g;
#endif
  }
}

// --------------------------- prep: build [A1n; B1n] -------------------------
__global__ __launch_bounds__(256) void prep_kernel(const float* __restrict__ A1B1,
                                                   float* __restrict__ M) {
  __shared__ float    s_sum[256];
  __shared__ float    s_max[256];
  __shared__ unsigned s_nz[256];
  const int b = blockIdx.x;
  const int t = threadIdx.x;
  const float* src = A1B1 + (size_t)b * 2048;

  float lsum = 0.f, lmax = 0.f;
  unsigned lnz = 0;
#pragma unroll
  for (int i = 0; i < 4; ++i) {
    const float v = fabsf(src[t + 256 * i]);
    lsum += v;
    lmax = fmaxf(lmax, v);
    lnz |= (src[1024 + t + 256 * i] != 0.f) ? 1u : 0u;
  }
  s_sum[t] = lsum; s_max[t] = lmax; s_nz[t] = lnz;
  __syncthreads();
  for (int off = 128; off > 0; off >>= 1) {
    if (t < off) {
      s_sum[t] += s_sum[t + off];
      s_max[t] = fmaxf(s_max[t], s_max[t + off]);
      s_nz[t] |= s_nz[t + off];
    }
    __syncthreads();
  }
  const float mean  = s_sum[0] * (1.0f / 1024.0f);
  const float mx    = s_max[0];
  const bool  reset = (mean > 1.0f) && (s_nz[0] == 0u);

  float* dst = M + (size_t)b * 2048;
#pragma unroll
  for (int i = 0; i < 4; ++i) {
    const int idx = t + 256 * i;
    dst[idx]        = reset ? (src[idx] / mx) : src[idx];        // A1n
    dst[1024 + idx] = reset ? 0.f : src[1024 + idx];             // B1n
  }
}

// ------------------------------ main GEMM -----------------------------------
__global__ __launch_bounds__(256) void aug_matmul_kernel(const float* __restrict__ M,
                                                         const float* __restrict__ A2,
                                                         float* __restrict__ out,
                                                         int N) {
  __shared__ float lds[2][64 * LDS_STRIDE];

  const int t     = threadIdx.x;
  const int lane  = t & 31;
  const int wave  = t >> 5;
  const int mtile = wave >> 1;   // 0..3 -> output rows mtile*16..+15
  const int ntile = wave & 1;    // 0..1 -> output cols ntile*16..+15
  const int b     = blockIdx.y;
  const int n0    = blockIdx.x * NPB;
  const int iters = (N - n0 < NPB) ? (N - n0) : NPB;

  const int lrow = lane & 15;
  const int hi   = (lane & 16) ? 1 : 0;
  const int col  = ntile * 16 + lrow;

  // Constant A-operand strip (16 rows x K=32) of [A1n; B1n], held in regs.
  // f32 16x4 A layout: lanes 0-15 hold K={0,1}, lanes 16-31 hold K={2,3}.
  const float* Mb = M + (size_t)b * 2048 + (size_t)(mtile * 16 + lrow) * 32;
  v2f areg[8];
#pragma unroll
  for (int kk = 0; kk < 8; ++kk)
    areg[kk] = *(const v2f*)(Mb + kk * 4 + hi * 2);

  const float* gbase = A2 + ((size_t)b * N + n0) * 2048;

  stage_tile(lds[0], gbase, t);

  for (int i = 0; i < iters; ++i) {
    if (i + 1 < iters) {
      stage_tile(lds[(i + 1) & 1], gbase + (size_t)(i + 1) * 2048, t);
      WAIT_ASYNC(2);   // keep next tile in flight, current tile complete
    } else {
      WAIT_ASYNC(0);
    }
    __syncthreads();

    const float* X = lds[i & 1];

    // C init: zeros for top tiles, Y (bottom half of input tile) for bottom.
    v8f c = {0.f, 0.f, 0.f, 0.f, 0.f, 0.f, 0.f, 0.f};
    if (mtile >= 2) {
      const float* Y = X + (32 + (mtile - 2) * 16 + hi * 8) * LDS_STRIDE + col;
#pragma unroll
      for (int r = 0; r < 8; ++r) c[r] = Y[r * LDS_STRIDE];
    }

    // K=32 as 8 x WMMA f32 16x16x4. B 4x16 layout: V0={K0|K2}, V1={K1|K3}.
#pragma unroll
    for (int kk = 0; kk < 8; ++kk) {
      const float* Bp = X + (kk * 4 + hi * 2) * LDS_STRIDE + col;
      v2f bb;
      bb.x = Bp[0];
      bb.y = Bp[LDS_STRIDE];
      c = __builtin_amdgcn_wmma_f32_16x16x4_f32(false, areg[kk], false, bb,
                                                (short)0, c, false, false);
    }

    float* od = out + (((size_t)b * N + (size_t)(n0 + i)) * 64 +
                       (size_t)(mtile * 16 + hi * 8)) * 32 + col;
#pragma unroll
    for (int r = 0; r < 8; ++r) od[(size_t)r * 32] = c[r];

    __syncthreads();   // protect lds[i&1] before it is restaged at i+1
  }
}

// ------------------------------ launcher ------------------------------------
extern "C" void kernel_launch(void* const* d_in, const int* in_sizes, int n_in,
                              void* d_out, int out_size, void* d_ws, size_t ws_size,
                              hipStream_t stream) {
  (void)n_in; (void)out_size; (void)ws_size;
  const float* A1B1 = (const float*)d_in[0];   // (B,1,64,32) f32
  const float* A2   = (const float*)d_in[1];   // (B,N,64,32) f32
  float* out = (float*)d_out;                  // (B,N,64,32) f32
  float* M   = (float*)d_ws;                   // (B,64,32) resolved [A1n;B1n]

  const int B = in_sizes[0] / 2048;            // 64*32 per batch
  const int N = in_sizes[1] / in_sizes[0];     // tiles per batch

  prep_kernel<<<dim3(B), dim3(256), 0, stream>>>(A1B1, M);

  dim3 grid((N + NPB - 1) / NPB, B);
  aug_matmul_kernel<<<grid, dim3(256), 0, stream>>>(M, A2, out, N);
}